// AttentiveDecoder_4277787427688
// MI455X (gfx1250) — compile-verified
//
#include <hip/hip_runtime.h>
#include <hip/hip_bf16.h>
#include <math.h>

#define B_   128
#define P_   49
#define D_   2048
#define H_   512
#define A_   512
#define E_   512
#define V_   15000
#define T_   21
#define TDEC 20
#define EOS_ 2

typedef float v2f __attribute__((ext_vector_type(2)));
typedef float v8f __attribute__((ext_vector_type(8)));

__device__ __forceinline__ float sigmoidf_(float x) { return 1.0f / (1.0f + expf(-x)); }

// ---------- setup: lengths, stable descending argsort, enc row map ----------
__global__ void k_sort(const int* __restrict__ captions,
                       int* __restrict__ len_s, int* __restrict__ sort_ind,
                       int* __restrict__ rowmap,
                       float* __restrict__ out_lens, float* __restrict__ out_sort) {
  __shared__ int slen[B_];
  int i = threadIdx.x;
  int l = 0; bool found = false;
  for (int t = 0; t < T_; ++t) {
    if (!found && captions[i * T_ + t] == EOS_) { l = t; found = true; }
  }
  slen[i] = l;
  __syncthreads();
  int r = 0;
  for (int j = 0; j < B_; ++j) {
    int lj = slen[j];
    r += (lj > l) || (lj == l && j < i);
  }
  len_s[r] = l;
  sort_ind[r] = i;
  out_lens[r] = (float)l;
  out_sort[r] = (float)i;
  // sorted row (b=r, pixel p) of encoder_out lives at original row i*P_+p
  for (int p = 0; p < P_; ++p) rowmap[r * P_ + p] = i * P_ + p;
}

__global__ void k_caps(const int* __restrict__ captions, const int* __restrict__ sort_ind,
                       int* __restrict__ caps_s, float* __restrict__ out_caps) {
  int idx = blockIdx.x * blockDim.x + threadIdx.x;
  if (idx >= B_ * T_) return;
  int b = idx / T_, j = idx % T_;
  int v = captions[sort_ind[b] * T_ + j];
  caps_s[idx] = v;
  out_caps[idx] = (float)v;
}

__global__ void k_mean(const float* __restrict__ enc, const int* __restrict__ sort_ind,
                       float* __restrict__ mean) {
  int idx = blockIdx.x * blockDim.x + threadIdx.x;
  if (idx >= B_ * D_) return;
  int b = idx / D_, d = idx % D_;
  const float* base = enc + ((size_t)sort_ind[b] * P_) * D_ + d;
  float s = 0.f;
  for (int p = 0; p < P_; ++p) s += base[(size_t)p * D_];
  mean[idx] = s * (1.0f / (float)P_);
}

// ---------- generic fp32 WMMA GEMM: C = A@B (+bias) (+C) (*mask) ----------
// 4 waves/block; each wave owns TWO 16x16 C tiles sharing one A fragment
// (block tile 16 x 128). No exec-divergent loads: OOB columns are clamped
// (garbage accumulates in clamped lanes but is never stored). The K loop is
// software-pipelined one 4-deep fragment ahead so VMEM loads for k+4 overlap
// the WMMAs consuming k (avoids s_wait_loadcnt 0 full drains).
// A-frag (16x4 f32): lane%16 -> M; VGPR j: K = j + 2*(lane>=16)
// B-frag (4x16 f32): lane%16 -> N; same K mapping
// C/D  (16x16 f32): VGPR r -> M = r + 8*(lane>=16), N = lane%16
__global__ void k_gemm(const float* __restrict__ A, int lda,
                       const int* __restrict__ rowmap,
                       const float* __restrict__ Bm, int ldb,
                       float* __restrict__ C, int ldc,
                       const float* __restrict__ bias,
                       int M, int N, int K, int accum,
                       const int* __restrict__ rowlen, int tstep) {
  const int lane = threadIdx.x & 31;
  const int wave = threadIdx.x >> 5;
  const int tm = blockIdx.y * 16;
  const int tn = (blockIdx.x * 4 + wave) * 32;
  const int half = lane >> 4;       // 0 or 1
  const int l16  = lane & 15;

  int m = tm + l16;
  if (m >= M) m = M - 1;            // clamp (value never stored)
  const int arow = rowmap ? rowmap[m] : m;

  const int n0 = tn + l16;
  const int n1 = n0 + 16;
  const int nc0 = (n0 < N) ? n0 : (N - 1);   // clamp: loads always legal
  const int nc1 = (n1 < N) ? n1 : (N - 1);

  const float* ap  = A  + (size_t)arow * lda + 2 * half;
  const float* bp0 = Bm + (size_t)(2 * half) * ldb + nc0;
  const float* bp1 = Bm + (size_t)(2 * half) * ldb + nc1;
  const size_t bstep = 4 * (size_t)ldb;

  v8f acc0 = {};
  v8f acc1 = {};

  // prologue: fragment for k = 0
  v2f a_c, b0_c, b1_c;
  a_c.x  = ap[0];
  a_c.y  = ap[1];
  b0_c.x = bp0[0];
  b0_c.y = bp0[(size_t)ldb];
  b1_c.x = bp1[0];
  b1_c.y = bp1[(size_t)ldb];

#pragma unroll 4
  for (int k = 0; k < K - 4; k += 4) {
    ap  += 4;
    bp0 += bstep;
    bp1 += bstep;
    v2f a_n, b0_n, b1_n;            // issue next fragment's loads first
    a_n.x  = ap[0];
    a_n.y  = ap[1];
    b0_n.x = bp0[0];
    b0_n.y = bp0[(size_t)ldb];
    b1_n.x = bp1[0];
    b1_n.y = bp1[(size_t)ldb];
    acc0 = __builtin_amdgcn_wmma_f32_16x16x4_f32(false, a_c, false, b0_c,
                                                 (short)0, acc0, false, false);
    acc1 = __builtin_amdgcn_wmma_f32_16x16x4_f32(false, a_c, false, b1_c,
                                                 (short)0, acc1, false, false);
    a_c = a_n; b0_c = b0_n; b1_c = b1_n;
  }
  acc0 = __builtin_amdgcn_wmma_f32_16x16x4_f32(false, a_c, false, b0_c,
                                               (short)0, acc0, false, false);
  acc1 = __builtin_amdgcn_wmma_f32_16x16x4_f32(false, a_c, false, b1_c,
                                               (short)0, acc1, false, false);

  const float bv0 = bias ? bias[nc0] : 0.0f;
  const float bv1 = bias ? bias[nc1] : 0.0f;
#pragma unroll
  for (int r = 0; r < 8; ++r) {
    int mr = tm + r + 8 * half;
    if (mr >= M) continue;
    float sc = rowlen ? ((tstep < rowlen[mr]) ? 1.0f : 0.0f) : 1.0f;
    if (n0 < N) {
      size_t ci = (size_t)mr * ldc + n0;
      float v = acc0[r] + bv0;
      if (accum) v += C[ci];
      C[ci] = v * sc;
    }
    if (n1 < N) {
      size_t ci = (size_t)mr * ldc + n1;
      float v = acc1[r] + bv1;
      if (accum) v += C[ci];
      C[ci] = v * sc;
    }
  }
}

// ---------- attention scores: e[b,p] = sum_a tanh(encp+decp)*fw + fb ----------
__global__ void k_scores(const float* __restrict__ encp, const float* __restrict__ decp,
                         const float* __restrict__ fw, const float* __restrict__ fb,
                         float* __restrict__ e) {
  int row = blockIdx.x * 4 + (threadIdx.x >> 5);
  int lane = threadIdx.x & 31;
  if (row >= B_ * P_) return;
  int b = row / P_;
  const float* ep = encp + (size_t)row * A_;
  const float* dp = decp + (size_t)b * A_;
  float s = 0.f;
  for (int a = lane; a < A_; a += 32) s += tanhf(ep[a] + dp[a]) * fw[a];
  for (int off = 16; off > 0; off >>= 1) s += __shfl_xor(s, off, 32);
  if (lane == 0) e[row] = s + fb[0];
}

// ---------- softmax over P=49, one wave per batch row; writes masked alphas ----------
__global__ void k_softmax(const float* __restrict__ e, float* __restrict__ alpha,
                          float* __restrict__ out_alphas,
                          const int* __restrict__ len_s, int t) {
  int b = blockIdx.x * 4 + (threadIdx.x >> 5);
  int lane = threadIdx.x & 31;
  const float* eb = e + (size_t)b * P_;
  float v0 = (lane < P_)      ? eb[lane]      : -INFINITY;
  float v1 = (lane + 32 < P_) ? eb[lane + 32] : -INFINITY;
  float mx = fmaxf(v0, v1);
  for (int off = 16; off > 0; off >>= 1) mx = fmaxf(mx, __shfl_xor(mx, off, 32));
  float e0 = (lane < P_)      ? expf(v0 - mx) : 0.f;
  float e1 = (lane + 32 < P_) ? expf(v1 - mx) : 0.f;
  float sm = e0 + e1;
  for (int off = 16; off > 0; off >>= 1) sm += __shfl_xor(sm, off, 32);
  float inv = 1.0f / sm;
  float maskf = (t < len_s[b]) ? 1.0f : 0.0f;
  size_t ob = ((size_t)b * TDEC + t) * P_;
  if (lane < P_) {
    alpha[b * P_ + lane] = e0 * inv;
    out_alphas[ob + lane] = e0 * inv * maskf;
  }
  if (lane + 32 < P_) {
    alpha[b * P_ + lane + 32] = e1 * inv;
    out_alphas[ob + lane + 32] = e1 * inv * maskf;
  }
}

// ---------- awe[b,d] = sum_p alpha[b,p] * enc_sorted[b,p,d] ----------
__global__ void k_awe(const float* __restrict__ enc, const int* __restrict__ sort_ind,
                      const float* __restrict__ alpha, float* __restrict__ awe) {
  int idx = blockIdx.x * blockDim.x + threadIdx.x;
  if (idx >= B_ * D_) return;
  int b = idx / D_, d = idx % D_;
  const float* base = enc + ((size_t)sort_ind[b] * P_) * D_ + d;
  const float* al = alpha + b * P_;
  float s = 0.f;
  for (int p = 0; p < P_; ++p) s += al[p] * base[(size_t)p * D_];
  awe[idx] = s;
}

// ---------- x = concat(emb[caps[:,t]], sigmoid(gbuf)*awe) ----------
__global__ void k_xbuild(const float* __restrict__ emb, const int* __restrict__ caps_s,
                         const float* __restrict__ gbuf, const float* __restrict__ awe,
                         float* __restrict__ x, int t) {
  const int XW = E_ + D_;
  int idx = blockIdx.x * blockDim.x + threadIdx.x;
  if (idx >= B_ * XW) return;
  int b = idx / XW, c = idx % XW;
  float v;
  if (c < E_) {
    int tok = caps_s[b * T_ + t];
    v = emb[(size_t)tok * E_ + c];
  } else {
    int d = c - E_;
    v = sigmoidf_(gbuf[b * D_ + d]) * awe[b * D_ + d];
  }
  x[idx] = v;
}

// ---------- LSTM cell (gate order i,f,g,o), h/c updated in place ----------
__global__ void k_lstm(const float* __restrict__ gates,
                       float* __restrict__ h, float* __restrict__ c) {
  int idx = blockIdx.x * blockDim.x + threadIdx.x;
  if (idx >= B_ * H_) return;
  int b = idx / H_, j = idx % H_;
  const float* g4 = gates + (size_t)b * 4 * H_;
  float ig = sigmoidf_(g4[j]);
  float fg = sigmoidf_(g4[H_ + j]);
  float gg = tanhf(g4[2 * H_ + j]);
  float og = sigmoidf_(g4[3 * H_ + j]);
  float cn = fg * c[idx] + ig * gg;
  float hn = og * tanhf(cn);
  c[idx] = cn;
  h[idx] = hn;
}

extern "C" void kernel_launch(void* const* d_in, const int* in_sizes, int n_in,
                              void* d_out, int out_size, void* d_ws, size_t ws_size,
                              hipStream_t stream) {
  (void)in_sizes; (void)n_in; (void)out_size; (void)ws_size;

  const float* enc        = (const float*)d_in[0];
  const int*   captions   = (const int*)  d_in[1];
  const float* emb        = (const float*)d_in[2];
  const float* W_ih       = (const float*)d_in[3];
  const float* W_hh       = (const float*)d_in[4];
  const float* b_lstm     = (const float*)d_in[5];
  const float* enc_att_w  = (const float*)d_in[6];
  const float* enc_att_b  = (const float*)d_in[7];
  const float* dec_att_w  = (const float*)d_in[8];
  const float* dec_att_b  = (const float*)d_in[9];
  const float* full_att_w = (const float*)d_in[10];
  const float* full_att_b = (const float*)d_in[11];
  const float* init_h_w   = (const float*)d_in[12];
  const float* init_h_b   = (const float*)d_in[13];
  const float* init_c_w   = (const float*)d_in[14];
  const float* init_c_b   = (const float*)d_in[15];
  const float* f_beta_w   = (const float*)d_in[16];
  const float* f_beta_b   = (const float*)d_in[17];
  const float* out_w      = (const float*)d_in[18];
  const float* out_b      = (const float*)d_in[19];

  // ---- output layout: preds | caps | lengths | alphas | sort_ind (all float) ----
  float* out     = (float*)d_out;
  float* o_preds = out;
  float* o_caps  = out + (size_t)B_ * TDEC * V_;
  float* o_lens  = o_caps + (size_t)B_ * T_;
  float* o_alph  = o_lens + B_;
  float* o_sort  = o_alph + (size_t)B_ * TDEC * P_;

  // ---- workspace carve-out (~20 MB) ----
  char* wsb = (char*)d_ws;
  size_t off = 0;
  auto carve = [&](size_t bytes) -> void* {
    void* p = wsb + off;
    off += (bytes + 255) & ~(size_t)255;
    return p;
  };
  int*   w_len    = (int*)  carve((size_t)B_ * 4);
  int*   w_sort   = (int*)  carve((size_t)B_ * 4);
  int*   w_caps   = (int*)  carve((size_t)B_ * T_ * 4);
  int*   w_rowmap = (int*)  carve((size_t)B_ * P_ * 4);
  float* w_mean   = (float*)carve((size_t)B_ * D_ * 4);
  float* w_encp   = (float*)carve((size_t)B_ * P_ * A_ * 4);
  float* w_h      = (float*)carve((size_t)B_ * H_ * 4);
  float* w_c      = (float*)carve((size_t)B_ * H_ * 4);
  float* w_decp   = (float*)carve((size_t)B_ * A_ * 4);
  float* w_e      = (float*)carve((size_t)B_ * P_ * 4);
  float* w_alpha  = (float*)carve((size_t)B_ * P_ * 4);
  float* w_awe    = (float*)carve((size_t)B_ * D_ * 4);
  float* w_g      = (float*)carve((size_t)B_ * D_ * 4);
  float* w_x      = (float*)carve((size_t)B_ * (E_ + D_) * 4);
  float* w_gates  = (float*)carve((size_t)B_ * 4 * H_ * 4);

  auto gemm = [&](const float* A, int lda, const int* rm,
                  const float* Bm, int ldb, float* C, int ldc,
                  const float* bias, int M, int N, int K,
                  int accum, const int* rl, int t) {
    dim3 g((N + 127) / 128, (M + 15) / 16);
    hipLaunchKernelGGL(k_gemm, g, dim3(128), 0, stream,
                       A, lda, rm, Bm, ldb, C, ldc, bias, M, N, K, accum, rl, t);
  };

  // ---- setup ----
  hipLaunchKernelGGL(k_sort, dim3(1), dim3(B_), 0, stream,
                     captions, w_len, w_sort, w_rowmap, o_lens, o_sort);
  hipLaunchKernelGGL(k_caps, dim3((B_ * T_ + 255) / 256), dim3(256), 0, stream,
                     captions, w_sort, w_caps, o_caps);
  hipLaunchKernelGGL(k_mean, dim3((B_ * D_ + 255) / 256), dim3(256), 0, stream,
                     enc, w_sort, w_mean);
  gemm(w_mean, D_, nullptr, init_h_w, H_, w_h, H_, init_h_b, B_, H_, D_, 0, nullptr, 0);
  gemm(w_mean, D_, nullptr, init_c_w, H_, w_c, H_, init_c_b, B_, H_, D_, 0, nullptr, 0);
  // enc_proj = enc_sorted @ enc_att_w + b  (gather fused via rowmap)
  gemm(enc, D_, w_rowmap, enc_att_w, A_, w_encp, A_, enc_att_b,
       B_ * P_, A_, D_, 0, nullptr, 0);

  // ---- 20-step sequential decode ----
  for (int t = 0; t < TDEC; ++t) {
    gemm(w_h, H_, nullptr, dec_att_w, A_, w_decp, A_, dec_att_b,
         B_, A_, H_, 0, nullptr, 0);
    hipLaunchKernelGGL(k_scores, dim3(B_ * P_ / 4), dim3(128), 0, stream,
                       w_encp, w_decp, full_att_w, full_att_b, w_e);
    hipLaunchKernelGGL(k_softmax, dim3(B_ / 4), dim3(128), 0, stream,
                       w_e, w_alpha, o_alph, w_len, t);
    hipLaunchKernelGGL(k_awe, dim3((B_ * D_ + 255) / 256), dim3(256), 0, stream,
                       enc, w_sort, w_alpha, w_awe);
    gemm(w_h, H_, nullptr, f_beta_w, D_, w_g, D_, f_beta_b,
         B_, D_, H_, 0, nullptr, 0);
    hipLaunchKernelGGL(k_xbuild, dim3((B_ * (E_ + D_) + 255) / 256), dim3(256), 0, stream,
                       emb, w_caps, w_g, w_awe, w_x, t);
    gemm(w_x, E_ + D_, nullptr, W_ih, 4 * H_, w_gates, 4 * H_, b_lstm,
         B_, 4 * H_, E_ + D_, 0, nullptr, 0);
    gemm(w_h, H_, nullptr, W_hh, 4 * H_, w_gates, 4 * H_, nullptr,
         B_, 4 * H_, H_, 1, nullptr, 0);
    hipLaunchKernelGGL(k_lstm, dim3((B_ * H_ + 255) / 256), dim3(256), 0, stream,
                       w_gates, w_h, w_c);
    // preds[b,t,:] = h_new @ out_w + out_b, masked by t < len_s[b]
    gemm(w_h, H_, nullptr, out_w, V_, o_preds + (size_t)t * V_, TDEC * V_, out_b,
         B_, V_, H_, 0, w_len, t);
  }
}